// BDFM_46428596469849
// MI455X (gfx1250) — compile-verified
//
#include <hip/hip_runtime.h>
#include <hip/hip_bf16.h>
#include <math.h>

// ---------------------------------------------------------------------------
// BDFM fused pipeline for MI455X (gfx1250, wave32).
//
// Shapes: B=8, C=256, H=W=128, HW=16384.
// The reference collapses exactly (BN is affine, mid has rank 3) to:
//     out[b,o,n] = sum_c Weff[b,o,c] * feature[b,c,n] + beff[b,o]
// The dominant kernel is a per-batch 256x256xHW fp32 GEMM on the matrix pipe
// (V_WMMA_F32_16X16X4_F32; fp32 kept: Weff magnitudes ~1e5 would overflow
// f16 and the workload is HBM-bound at ~384MB / 23.3TB/s ~= 16us, so lower
// precision buys nothing). Feature tiles are staged into LDS with
// GLOBAL_LOAD_ASYNC_TO_LDS_B128 (ASYNCcnt, double buffered) so each feature
// element crosses the L2->WGP link once per block instead of 2x.
// ---------------------------------------------------------------------------

typedef __attribute__((ext_vector_type(2))) float v2f;
typedef __attribute__((ext_vector_type(8))) float v8f;

#define BB 8
#define CC 256
#define HH 128
#define WW 128
#define HWN 16384
#define BN_EPS 1e-5f

// ---------------------------------------------------------------------------
// Kernel 1: fbu[b,{er,1-di,di-er},n] from m via threshold + 4x4 min/max pool.
// SAME padding for k=4,s=1: pad_lo=1, pad_hi=2; pad values are identity for
// min/max so we just clip the window.
// ---------------------------------------------------------------------------
__global__ __launch_bounds__(256)
void fbu_kernel(const float* __restrict__ m, float* __restrict__ fbu) {
    int idx = blockIdx.x * 256 + threadIdx.x;   // 0 .. B*HW-1
    int b = idx >> 14;
    int n = idx & (HWN - 1);
    int i = n >> 7;
    int j = n & (WW - 1);
    const float* mb = m + (size_t)b * HWN;
    float er = 1.f, di = 0.f;
    #pragma unroll
    for (int dr = -1; dr <= 2; ++dr) {
        int r = i + dr;
        if (r < 0 || r >= HH) continue;
        #pragma unroll
        for (int dc = -1; dc <= 2; ++dc) {
            int c = j + dc;
            if (c < 0 || c >= WW) continue;
            float z = (fmaxf(mb[r * WW + c], 0.f) > 0.3f) ? 1.f : 0.f;
            er = fminf(er, z);
            di = fmaxf(di, z);
        }
    }
    float* fb = fbu + (size_t)b * 3 * HWN;
    fb[n]           = er;
    fb[HWN + n]     = 1.f - di;
    fb[2 * HWN + n] = di - er;
}

// ---------------------------------------------------------------------------
// Kernel 2: mid[b,k,c] = sum_n fbu[b,k,n] * feature[b,c,n].
// ---------------------------------------------------------------------------
__global__ __launch_bounds__(256)
void mid_kernel(const float* __restrict__ feature, const float* __restrict__ fbu,
                float* __restrict__ mid) {
    int c = blockIdx.x, b = blockIdx.y, t = threadIdx.x;
    const float* f  = feature + ((size_t)b * CC + c) * HWN;
    const float* fb = fbu + (size_t)b * 3 * HWN;
    float s0 = 0.f, s1 = 0.f, s2 = 0.f;
    for (int n = t; n < HWN; n += 256) {
        float fv = f[n];
        s0 += fv * fb[n];
        s1 += fv * fb[HWN + n];
        s2 += fv * fb[2 * HWN + n];
    }
    __shared__ float red[3][256];
    red[0][t] = s0; red[1][t] = s1; red[2][t] = s2;
    __syncthreads();
    for (int s = 128; s > 0; s >>= 1) {
        if (t < s) {
            red[0][t] += red[0][t + s];
            red[1][t] += red[1][t + s];
            red[2][t] += red[2][t + s];
        }
        __syncthreads();
    }
    if (t == 0) {
        mid[((size_t)b * 3 + 0) * CC + c] = red[0][0];
        mid[((size_t)b * 3 + 1) * CC + c] = red[1][0];
        mid[((size_t)b * 3 + 2) * CC + c] = red[2][0];
    }
}

// ---------------------------------------------------------------------------
// Kernel 3a: H[b,k,c'] = sum_o mid[b,k,o]*scale_f[o]*w_feat[o,c'];
//            d[b,k]    = sum_o mid[b,k,o]*(beta_f[o]-mean_f[o]*scale_f[o]).
// ---------------------------------------------------------------------------
__global__ __launch_bounds__(256)
void hd_kernel(const float* __restrict__ mid, const float* __restrict__ w_feat,
               const float* __restrict__ gamma, const float* __restrict__ beta,
               const float* __restrict__ mean, const float* __restrict__ var,
               float* __restrict__ H, float* __restrict__ d) {
    int k = blockIdx.x, b = blockIdx.y, cp = threadIdx.x;
    const float* mk = mid + ((size_t)b * 3 + k) * CC;
    float acc = 0.f;
    for (int o = 0; o < CC; ++o) {
        float sf = gamma[o] * rsqrtf(var[o] + BN_EPS);
        acc += mk[o] * sf * w_feat[o * CC + cp];
    }
    H[((size_t)b * 3 + k) * CC + cp] = acc;
    if (cp == 0) {
        float dd = 0.f;
        for (int o = 0; o < CC; ++o) {
            float sf = gamma[o] * rsqrtf(var[o] + BN_EPS);
            dd += mk[o] * (beta[o] - mean[o] * sf);
        }
        d[b * 3 + k] = dd;
    }
}

// ---------------------------------------------------------------------------
// Kernel 3b: G[b,o,k] = sum_c w_out[o,256+c]*mid[b,k,c] (block reduction),
//   Weff[b,o,c] = scale_o[o]*(w_out[o,c] + sum_k G_k*H[b,k,c]),
//   beff[b,o]   = (beta_o-mean_o*scale_o)[o] + scale_o[o]*sum_k G_k*d[b,k].
// ---------------------------------------------------------------------------
__global__ __launch_bounds__(256)
void weff_kernel(const float* __restrict__ mid, const float* __restrict__ H,
                 const float* __restrict__ dv, const float* __restrict__ w_out,
                 const float* __restrict__ gamma, const float* __restrict__ beta,
                 const float* __restrict__ mean, const float* __restrict__ var,
                 float* __restrict__ Weff, float* __restrict__ beff) {
    int o = blockIdx.x, b = blockIdx.y, c = threadIdx.x;
    __shared__ float red[3][256];
    float w2 = w_out[o * 512 + 256 + c];
    red[0][c] = w2 * mid[((size_t)b * 3 + 0) * CC + c];
    red[1][c] = w2 * mid[((size_t)b * 3 + 1) * CC + c];
    red[2][c] = w2 * mid[((size_t)b * 3 + 2) * CC + c];
    __syncthreads();
    for (int s = 128; s > 0; s >>= 1) {
        if (c < s) {
            red[0][c] += red[0][c + s];
            red[1][c] += red[1][c + s];
            red[2][c] += red[2][c + s];
        }
        __syncthreads();
    }
    float G0 = red[0][0], G1 = red[1][0], G2 = red[2][0];
    float so = gamma[o] * rsqrtf(var[o] + BN_EPS);
    const float* Hb = H + (size_t)b * 3 * CC;
    Weff[(((size_t)b * CC) + o) * CC + c] =
        so * (w_out[o * 512 + c] + G0 * Hb[c] + G1 * Hb[CC + c] + G2 * Hb[2 * CC + c]);
    if (c == 0) {
        beff[b * CC + o] = (beta[o] - mean[o] * so) +
                           so * (G0 * dv[b * 3 + 0] + G1 * dv[b * 3 + 1] + G2 * dv[b * 3 + 2]);
    }
}

// ---------------------------------------------------------------------------
// Kernel 4: batched fp32 WMMA GEMM with async-to-LDS double buffering.
//   out[b,o,n] = sum_c Weff[b,o,c]*feature[b,c,n] + beff[b,o]
//
// Block: 256 threads = 8 waves, output 128(M) x 128(N). Waves: 2 along M
// (wm, 64 rows = 4 M-tiles) x 4 along N (wn, 32 cols = 2 N-tiles) so every
// feature element is consumed from LDS, not refetched per wave.
//
// K loop: chunks of 16 rows. Each chunk [16 x 128] of feature is pulled by
// GLOBAL_LOAD_ASYNC_TO_LDS_B128 (two b128 per thread), ASYNCcnt-tracked,
// double-buffered against the 32 WMMAs of the current chunk.
//
// LDS tile row stride = 144 floats: 2*144 mod 64 = 32 dwords, so the two
// half-wave phases (rows kk / kk+2) hit disjoint bank ranges -> conflict-free
// ds_load_b32 consumption.
//
// WMMA V_WMMA_F32_16X16X4_F32 VGPR layouts (CDNA5 ISA):
//  A 16x4: lane m=(l&15), khalf=(l>>4); VGPR v holds K=2*khalf+v  -> float2
//          from row-major Weff.
//  B 4x16: lane n=(l&15), khalf=(l>>4); VGPR v holds K=2*khalf+v  -> two LDS
//          reads at rows krow, krow+1.
//  C/D:    VGPR r -> row M=r+8*khalf, col N=l&15.
// Exact tiling + uniform branches keep EXEC all-ones as WMMA requires.
// ---------------------------------------------------------------------------
__global__ __launch_bounds__(256)
void gemm_out_kernel(const float* __restrict__ feature, const float* __restrict__ Weff,
                     const float* __restrict__ beff, float* __restrict__ out) {
    const int b    = blockIdx.z;
    const int lane = threadIdx.x & 31;
    const int wave = threadIdx.x >> 5;
    const int wm = wave & 1;                    // 2 waves along M
    const int wn = wave >> 1;                   // 4 waves along N
    const int m_base = blockIdx.y * 128 + wm * 64;
    const int n_blk  = blockIdx.x * 128;
    const int n_off  = wn * 32;

    const float* W  = Weff + (size_t)b * CC * CC;
    const float* F  = feature + (size_t)b * CC * HWN;
    const float* Bf = beff + (size_t)b * CC;
    float* O = out + (size_t)b * CC * HWN;

    const int l15   = lane & 15;
    const int khalf = lane >> 4;                // 0 or 1

    // [buf][16 K-rows x 128 N, row stride padded to 144 floats]
    __shared__ float tile[2][16 * 144];

    // --- async producer setup: thread t loads rows (t>>5) and (t>>5)+8,
    //     16-byte segment seg = t&31 (32 segs * 16B = 512B = 128 floats). ---
    const int t   = threadIdx.x;
    const int row = t >> 5;                     // 0..7
    const int seg = t & 31;
    const float* g0 = F + (size_t)row * HWN + n_blk + seg * 4;
    const float* g1 = F + (size_t)(row + 8) * HWN + n_blk + seg * 4;
    // Generic->LDS: low 32 bits of a generic LDS address are the wave-relative
    // LDS byte offset (aperture truncation, ISA 10.2).
    const unsigned lds0 = (unsigned)(size_t)&tile[0][row * 144 + seg * 4];
    const unsigned lds1 = (unsigned)(size_t)&tile[0][(row + 8) * 144 + seg * 4];
    const unsigned bufstride = 16u * 144u * 4u;

    #define ISSUE_CHUNK(k0, bufsel)                                             \
        do {                                                                     \
            const float* _a0 = g0 + (size_t)(k0) * HWN;                          \
            const float* _a1 = g1 + (size_t)(k0) * HWN;                          \
            unsigned _d0 = lds0 + (unsigned)(bufsel) * bufstride;                \
            unsigned _d1 = lds1 + (unsigned)(bufsel) * bufstride;                \
            asm volatile("global_load_async_to_lds_b128 %0, %1, off"             \
                         :: "v"(_d0), "v"(_a0) : "memory");                      \
            asm volatile("global_load_async_to_lds_b128 %0, %1, off"             \
                         :: "v"(_d1), "v"(_a1) : "memory");                      \
        } while (0)

    // --- accumulators + A row pointers ---
    v8f zero;
    #pragma unroll
    for (int e = 0; e < 8; ++e) zero[e] = 0.f;
    v8f acc[4][2];
    #pragma unroll
    for (int mt = 0; mt < 4; ++mt)
        #pragma unroll
        for (int nt = 0; nt < 2; ++nt) acc[mt][nt] = zero;

    const float* Wr[4];
    #pragma unroll
    for (int mt = 0; mt < 4; ++mt)
        Wr[mt] = W + (size_t)(m_base + mt * 16 + l15) * CC + 2 * khalf;

    // --- prologue: chunk 0 ---
    ISSUE_CHUNK(0, 0);
    asm volatile("s_wait_asynccnt 0x0" ::: "memory");
    __syncthreads();

    // --- main K loop: 16 chunks of 16 ---
    for (int kc = 0; kc < 16; ++kc) {
        if (kc + 1 < 16) ISSUE_CHUNK((kc + 1) * 16, (kc + 1) & 1);

        const float* lb = &tile[kc & 1][0];
        #pragma unroll
        for (int kk4 = 0; kk4 < 16; kk4 += 4) {
            const int kglob = kc * 16 + kk4;
            v2f a[4];
            #pragma unroll
            for (int mt = 0; mt < 4; ++mt)
                a[mt] = *(const v2f*)(Wr[mt] + kglob);

            const int krow = kk4 + 2 * khalf;
            v2f bf2[2];
            #pragma unroll
            for (int nt = 0; nt < 2; ++nt) {
                const int ci = krow * 144 + n_off + nt * 16 + l15;
                bf2[nt][0] = lb[ci];
                bf2[nt][1] = lb[ci + 144];
            }
            #pragma unroll
            for (int mt = 0; mt < 4; ++mt)
                #pragma unroll
                for (int nt = 0; nt < 2; ++nt)
                    acc[mt][nt] = __builtin_amdgcn_wmma_f32_16x16x4_f32(
                        false, a[mt], false, bf2[nt], (short)0, acc[mt][nt],
                        false, false);
        }

        asm volatile("s_wait_asynccnt 0x0" ::: "memory");
        __syncthreads();
    }
    #undef ISSUE_CHUNK

    // --- epilogue: add beff, store ---
    #pragma unroll
    for (int mt = 0; mt < 4; ++mt) {
        #pragma unroll
        for (int r = 0; r < 8; ++r) {
            const int o = m_base + mt * 16 + r + 8 * khalf;
            const float bias = Bf[o];
            float* orow = O + (size_t)o * HWN;
            #pragma unroll
            for (int nt = 0; nt < 2; ++nt)
                orow[n_blk + n_off + nt * 16 + l15] = acc[mt][nt][r] + bias;
        }
    }
}

// ---------------------------------------------------------------------------
// Launch. Workspace layout (floats), total ~3.73 MB:
//   fbu[B,3,HW] | mid[B,3,C] | H[B,3,C] | d[B,3](pad 32) | beff[B,C] | Weff[B,C,C]
// ---------------------------------------------------------------------------
extern "C" void kernel_launch(void* const* d_in, const int* in_sizes, int n_in,
                              void* d_out, int out_size, void* d_ws, size_t ws_size,
                              hipStream_t stream) {
    const float* feature   = (const float*)d_in[0];
    const float* m         = (const float*)d_in[1];
    const float* w_feat    = (const float*)d_in[2];
    const float* bn_f_g    = (const float*)d_in[3];
    const float* bn_f_b    = (const float*)d_in[4];
    const float* bn_f_m    = (const float*)d_in[5];
    const float* bn_f_v    = (const float*)d_in[6];
    const float* w_out     = (const float*)d_in[7];
    const float* bn_o_g    = (const float*)d_in[8];
    const float* bn_o_b    = (const float*)d_in[9];
    const float* bn_o_m    = (const float*)d_in[10];
    const float* bn_o_v    = (const float*)d_in[11];
    float* out = (float*)d_out;

    float* ws   = (float*)d_ws;
    float* fbu  = ws;
    float* mid  = fbu + (size_t)BB * 3 * HWN;
    float* H    = mid + (size_t)BB * 3 * CC;
    float* dv   = H   + (size_t)BB * 3 * CC;
    float* beff = dv  + 32;
    float* Weff = beff + (size_t)BB * CC;

    fbu_kernel<<<dim3((BB * HWN) / 256), dim3(256), 0, stream>>>(m, fbu);
    mid_kernel<<<dim3(CC, BB), dim3(256), 0, stream>>>(feature, fbu, mid);
    hd_kernel<<<dim3(3, BB), dim3(256), 0, stream>>>(mid, w_feat, bn_f_g, bn_f_b,
                                                     bn_f_m, bn_f_v, H, dv);
    weff_kernel<<<dim3(CC, BB), dim3(256), 0, stream>>>(mid, H, dv, w_out,
                                                        bn_o_g, bn_o_b, bn_o_m, bn_o_v,
                                                        Weff, beff);
    gemm_out_kernel<<<dim3(HWN / 128, CC / 128, BB), dim3(256), 0, stream>>>(
        feature, Weff, beff, out);
}